// GraphSAGE_28037546508931
// MI455X (gfx1250) — compile-verified
//
#include <hip/hip_runtime.h>

#define N_NODES 20000
#define N_EDGES 640000
#define HID 128
#define NEG_SLOPE 0.1f

typedef __attribute__((ext_vector_type(2))) float v2f;
typedef __attribute__((ext_vector_type(8))) float v8f;

// ---------------------------------------------------------------------------
// CSR construction
// ---------------------------------------------------------------------------
__global__ void sage_zero_i32(int* __restrict__ p, int n) {
    int i = blockIdx.x * blockDim.x + threadIdx.x;
    if (i < n) p[i] = 0;
}

__global__ void sage_count_deg(const int* __restrict__ edge_index,
                               int* __restrict__ deg) {
    int e = blockIdx.x * blockDim.x + threadIdx.x;
    if (e < N_EDGES) {
        int d = edge_index[N_EDGES + e];  // dst row of [2, E]
        atomicAdd(&deg[d], 1);
    }
}

// Single-block exclusive scan over 20000 degrees -> row_ptr, cursor, inv_deg.
__global__ __launch_bounds__(1024) void sage_scan_deg(
    const int* __restrict__ deg, int* __restrict__ row_ptr,
    int* __restrict__ cursor, float* __restrict__ inv_deg, int n) {
    __shared__ int sums[1024];
    const int t = threadIdx.x;
    const int CH = (n + 1023) / 1024;
    const int beg = t * CH;
    const int end = (beg + CH < n) ? (beg + CH) : n;
    int local = 0;
    for (int i = beg; i < end; ++i) local += deg[i];
    sums[t] = local;
    __syncthreads();
    // Hillis-Steele inclusive scan of per-thread partials
    for (int off = 1; off < 1024; off <<= 1) {
        int v = sums[t];
        int add = (t >= off) ? sums[t - off] : 0;
        __syncthreads();
        sums[t] = v + add;
        __syncthreads();
    }
    int run = (t == 0) ? 0 : sums[t - 1];
    for (int i = beg; i < end; ++i) {
        row_ptr[i] = run;
        cursor[i]  = run;
        int d = deg[i];
        inv_deg[i] = 1.0f / (float)((d > 1) ? d : 1);
        run += d;
    }
    if (t == 1023) row_ptr[n] = sums[1023];
}

__global__ void sage_fill_csr(const int* __restrict__ edge_index,
                              int* __restrict__ cursor,
                              int* __restrict__ csr_src) {
    int e = blockIdx.x * blockDim.x + threadIdx.x;
    if (e < N_EDGES) {
        int s = edge_index[e];
        int d = edge_index[N_EDGES + e];
        int pos = atomicAdd(&cursor[d], 1);
        csr_src[pos] = s;
    }
}

// ---------------------------------------------------------------------------
// Mean aggregation: one wave32 per node, lane owns a float4 of the 128-row.
// Coalesced 512B row reads (global_load_b128), L2-resident working set.
// ---------------------------------------------------------------------------
__global__ __launch_bounds__(256) void sage_aggregate(
    const float* __restrict__ h, const int* __restrict__ row_ptr,
    const int* __restrict__ csr_src, const float* __restrict__ inv_deg,
    float* __restrict__ agg) {
    const int wave = (blockIdx.x * blockDim.x + threadIdx.x) >> 5;
    const int lane = threadIdx.x & 31;
    if (wave >= N_NODES) return;
    const int beg = row_ptr[wave];
    const int end = row_ptr[wave + 1];
    float4 s = make_float4(0.f, 0.f, 0.f, 0.f);
    int nxt = (beg < end) ? csr_src[beg] : 0;
    for (int j = beg; j < end; ++j) {
        const int src = nxt;
        if (j + 1 < end) nxt = csr_src[j + 1];
        const float4 v = ((const float4*)(h + (size_t)src * HID))[lane];
        s.x += v.x; s.y += v.y; s.z += v.z; s.w += v.w;
    }
    const float id = inv_deg[wave];
    s.x *= id; s.y *= id; s.z *= id; s.w *= id;
    ((float4*)(agg + (size_t)wave * HID))[lane] = s;
}

// ---------------------------------------------------------------------------
// WMMA f32 16x16x4 GEMM helpers.
// A fragment (16x4 f32): lane<16 -> M=lane, K={k,k+1}; lane>=16 -> K={k+2,k+3}
// B fragment (4x16 f32): N across lanes (lane&15), K split by lane half
// C/D: 8 VGPRs, N=lane&15, M=(lane>>4)*8 + v
// ---------------------------------------------------------------------------
__device__ __forceinline__ v8f wmma_k128(const float* __restrict__ Arow,
                                         const float* __restrict__ wtile,
                                         int row, int half, v8f acc) {
    const float* Ak = Arow + half * 2;
#pragma unroll
    for (int k = 0; k < HID; k += 4) {
        v2f a = *(const v2f*)(Ak + k);
        int kb = k + half * 2;
        v2f b;
        b.x = wtile[kb * 16 + row];
        b.y = wtile[(kb + 1) * 16 + row];
        acc = __builtin_amdgcn_wmma_f32_16x16x4_f32(
            false, a, false, b, (short)0, acc, false, false);
    }
    return acc;
}

// out = leaky(A @ W + bias); A:[M,128], W:[128,128], block = 4 waves,
// each wave one 16x16 tile; W N-tile staged in LDS (shared by 4 waves).
__global__ __launch_bounds__(128) void sage_gemm_bias_leaky(
    const float* __restrict__ A, const float* __restrict__ W,
    const float* __restrict__ bias, float* __restrict__ out, int M) {
    __shared__ float wtile[HID * 16];
    const int lane = threadIdx.x;
    const int wv   = threadIdx.y;
    const int nt   = blockIdx.x;
    const int mt   = blockIdx.y * 4 + wv;
    {   // stage W[:, nt*16 .. nt*16+16) as [k][n]
        const int k = wv * 32 + lane;            // 0..127
        const float* wr = W + (size_t)k * HID + nt * 16;
#pragma unroll
        for (int j = 0; j < 16; ++j) wtile[k * 16 + j] = wr[j];
    }
    __syncthreads();
    if (mt * 16 >= M) return;

    const int row  = lane & 15;
    const int half = lane >> 4;
    const float* Arow = A + (size_t)(mt * 16 + row) * HID;
    v8f acc = {0.f, 0.f, 0.f, 0.f, 0.f, 0.f, 0.f, 0.f};
    acc = wmma_k128(Arow, wtile, row, half, acc);

    const float bcol = bias[nt * 16 + row];
    float* orow = out + (size_t)(mt * 16 + half * 8) * HID + nt * 16 + row;
#pragma unroll
    for (int v = 0; v < 8; ++v) {
        float c = acc[v] + bcol;
        orow[(size_t)v * HID] = (c >= 0.f) ? c : NEG_SLOPE * c;
    }
}

// out = leaky(A1 @ W1 + A2 @ W2 + bias)  (layer update: agg*Wl + h*Wr + b)
__global__ __launch_bounds__(128) void sage_gemm2_bias_leaky(
    const float* __restrict__ A1, const float* __restrict__ W1,
    const float* __restrict__ A2, const float* __restrict__ W2,
    const float* __restrict__ bias, float* __restrict__ out, int M) {
    __shared__ float wt1[HID * 16];
    __shared__ float wt2[HID * 16];
    const int lane = threadIdx.x;
    const int wv   = threadIdx.y;
    const int nt   = blockIdx.x;
    const int mt   = blockIdx.y * 4 + wv;
    {
        const int k = wv * 32 + lane;
        const float* w1r = W1 + (size_t)k * HID + nt * 16;
        const float* w2r = W2 + (size_t)k * HID + nt * 16;
#pragma unroll
        for (int j = 0; j < 16; ++j) wt1[k * 16 + j] = w1r[j];
#pragma unroll
        for (int j = 0; j < 16; ++j) wt2[k * 16 + j] = w2r[j];
    }
    __syncthreads();
    if (mt * 16 >= M) return;

    const int row  = lane & 15;
    const int half = lane >> 4;
    const float* A1row = A1 + (size_t)(mt * 16 + row) * HID;
    const float* A2row = A2 + (size_t)(mt * 16 + row) * HID;
    v8f acc = {0.f, 0.f, 0.f, 0.f, 0.f, 0.f, 0.f, 0.f};
    acc = wmma_k128(A1row, wt1, row, half, acc);
    acc = wmma_k128(A2row, wt2, row, half, acc);

    const float bcol = bias[nt * 16 + row];
    float* orow = out + (size_t)(mt * 16 + half * 8) * HID + nt * 16 + row;
#pragma unroll
    for (int v = 0; v < 8; ++v) {
        float c = acc[v] + bcol;
        orow[(size_t)v * HID] = (c >= 0.f) ? c : NEG_SLOPE * c;
    }
}

// ---------------------------------------------------------------------------
extern "C" void kernel_launch(void* const* d_in, const int* in_sizes, int n_in,
                              void* d_out, int out_size, void* d_ws, size_t ws_size,
                              hipStream_t stream) {
    const float* x        = (const float*)d_in[0];
    const float* enc_w    = (const float*)d_in[1];
    const float* enc_b    = (const float*)d_in[2];
    const float* lin_l_w  = (const float*)d_in[3];   // [2,128,128]
    const float* lin_l_b  = (const float*)d_in[4];   // [2,128]
    const float* lin_r_w  = (const float*)d_in[5];   // [2,128,128]
    const int*   edge_idx = (const int*)d_in[6];     // [2,E] (jax default x32)

    // workspace layout
    char* ws = (char*)d_ws;
    size_t off = 0;
    auto alloc = [&](size_t bytes) -> void* {
        void* p = ws + off;
        off = (off + bytes + 255) & ~(size_t)255;
        return p;
    };
    float* h_a     = (float*)alloc((size_t)N_NODES * HID * sizeof(float));
    float* h_b     = (float*)alloc((size_t)N_NODES * HID * sizeof(float));
    float* agg     = (float*)alloc((size_t)N_NODES * HID * sizeof(float));
    int*   deg     = (int*)  alloc((size_t)N_NODES * sizeof(int));
    int*   row_ptr = (int*)  alloc((size_t)(N_NODES + 1) * sizeof(int));
    int*   cursor  = (int*)  alloc((size_t)N_NODES * sizeof(int));
    float* inv_deg = (float*)alloc((size_t)N_NODES * sizeof(float));
    int*   csr_src = (int*)  alloc((size_t)N_EDGES * sizeof(int));

    // 1) CSR build
    sage_zero_i32<<<(N_NODES + 255) / 256, 256, 0, stream>>>(deg, N_NODES);
    sage_count_deg<<<(N_EDGES + 255) / 256, 256, 0, stream>>>(edge_idx, deg);
    sage_scan_deg<<<1, 1024, 0, stream>>>(deg, row_ptr, cursor, inv_deg, N_NODES);
    sage_fill_csr<<<(N_EDGES + 255) / 256, 256, 0, stream>>>(edge_idx, cursor, csr_src);

    const dim3 gblk(32, 4);
    const dim3 ggrd(HID / 16, (N_NODES / 16 + 3) / 4);  // (8, 313)
    const int  agg_blocks = (N_NODES * 32 + 255) / 256; // wave per node

    // 2) encoder: h_a = leaky(x @ enc_w + enc_b)
    sage_gemm_bias_leaky<<<ggrd, gblk, 0, stream>>>(x, enc_w, enc_b, h_a, N_NODES);

    // 3) layer 0
    sage_aggregate<<<agg_blocks, 256, 0, stream>>>(h_a, row_ptr, csr_src, inv_deg, agg);
    sage_gemm2_bias_leaky<<<ggrd, gblk, 0, stream>>>(
        agg, lin_l_w, h_a, lin_r_w, lin_l_b, h_b, N_NODES);

    // 4) layer 1 -> d_out
    sage_aggregate<<<agg_blocks, 256, 0, stream>>>(h_b, row_ptr, csr_src, inv_deg, agg);
    sage_gemm2_bias_leaky<<<ggrd, gblk, 0, stream>>>(
        agg, lin_l_w + HID * HID, h_b, lin_r_w + HID * HID, lin_l_b + HID,
        (float*)d_out, N_NODES);
}